// IOU_76639396429833
// MI455X (gfx1250) — compile-verified
//
#include <hip/hip_runtime.h>
#include <hip/hip_bf16.h>

typedef float v4f __attribute__((ext_vector_type(4)));

#define B_SIZE 32
#define NCH 3
#define THREADS 256
#define BLOCKS_PER_BATCH 128
#define NSTREAMS 6                                  // pred c0,c1,c2 + targ c0,c1,c2
#define STREAM_TILE_BYTES (THREADS * 16)            // 4 KB per stream per tile
#define BUF_BYTES (NSTREAMS * STREAM_TILE_BYTES)    // 24 KB per buffer

__global__ void iou_zero_ws(unsigned int* ws) {
    if (threadIdx.x < 2 * B_SIZE) ws[threadIdx.x] = 0u;
}

// Issue one per-lane async global->LDS 16B copy (CDNA5 async engine, ASYNCcnt).
// NT temporal hint: stream is touched once, 805MB >> 192MB L2.
__device__ __forceinline__ void async_b128(unsigned lds_addr, unsigned long long gaddr) {
    asm volatile("global_load_async_to_lds_b128 %0, %1, off th:TH_LOAD_NT"
                 :: "v"(lds_addr), "v"(gaddr) : "memory");
}

__global__ __launch_bounds__(THREADS)
void iou_count_async(const float* __restrict__ pred,
                     const float* __restrict__ targ,
                     unsigned int* __restrict__ interCnt,
                     unsigned int* __restrict__ unionCnt,
                     int q /* float4 elements per channel per batch */) {
    __shared__ v4f smem[2][NSTREAMS][THREADS];      // 48 KB double buffer

    const int b   = blockIdx.y;
    const int tid = threadIdx.x;

    const size_t base = (size_t)b * NCH * q;        // in float4 units
    const v4f* __restrict__ pb = (const v4f*)pred + base;
    const v4f* __restrict__ tb = (const v4f*)targ + base;

    const int chunk      = q / gridDim.x;           // float4 per block (2048)
    const int numTiles   = chunk / THREADS;         // tiles per block (8)
    const int blockStart = blockIdx.x * chunk;

    // Low 32 bits of the flat address of an LDS object = wave-relative LDS
    // byte offset (ISA 10.2), which is what async-load VDST consumes.
    const unsigned lds0 = (unsigned)(size_t)(&smem[0][0][0]);

    const v4f* gs[NSTREAMS] = { pb, pb + (size_t)q, pb + 2 * (size_t)q,
                                tb, tb + (size_t)q, tb + 2 * (size_t)q };

    auto issue_tile = [&](int t, int sel) {
        const int i = blockStart + t * THREADS + tid;
#pragma unroll
        for (int s = 0; s < NSTREAMS; ++s) {
            unsigned lds = lds0 + (unsigned)((sel * NSTREAMS + s) * STREAM_TILE_BYTES)
                                + (unsigned)tid * 16u;
            unsigned long long ga = (unsigned long long)(size_t)(gs[s] + i);
            async_b128(lds, ga);
        }
    };

    unsigned int inter = 0, uni = 0;

    issue_tile(0, 0);                               // prologue: 6 async ops in flight
    for (int t = 0; t < numTiles; ++t) {
        const int sel = t & 1;
        if (t + 1 < numTiles) {
            issue_tile(t + 1, sel ^ 1);             // up to 12 in flight
            // async loads complete in order: <=6 outstanding => tile t resident
            asm volatile("s_wait_asynccnt 0x6" ::: "memory");
        } else {
            asm volatile("s_wait_asynccnt 0x0" ::: "memory");
        }

        // Each thread consumes exactly the 6x16B it staged (no cross-wave
        // sharing => no barrier needed; ASYNCcnt wait is sufficient).
        v4f p0 = smem[sel][0][tid];
        v4f p1 = smem[sel][1][tid];
        v4f p2 = smem[sel][2][tid];
        v4f t0 = smem[sel][3][tid];
        v4f t1 = smem[sel][4][tid];
        v4f t2 = smem[sel][5][tid];

#pragma unroll
        for (int k = 0; k < 4; ++k) {
            // argmax over {c0,c1,c2} > 0  <=>  max(c1,c2) > c0
            unsigned int p = (p1[k] > p0[k]) || (p2[k] > p0[k]);
            unsigned int w = (t1[k] > t0[k]) || (t2[k] > t0[k]);
            inter += p & w;
            uni   += p | w;
        }

        // WAR guard: LDS reads of tile t must retire before the async engine
        // overwrites this buffer when tile t+2 is issued next iteration.
        asm volatile("s_wait_dscnt 0x0" ::: "memory");
    }

    // wave32 butterfly reduction
#pragma unroll
    for (int off = 16; off > 0; off >>= 1) {
        inter += __shfl_xor(inter, off, 32);
        uni   += __shfl_xor(uni,   off, 32);
    }

    __shared__ unsigned int sI[THREADS / 32];
    __shared__ unsigned int sU[THREADS / 32];
    const int lane = threadIdx.x & 31;
    const int wid  = threadIdx.x >> 5;
    if (lane == 0) { sI[wid] = inter; sU[wid] = uni; }
    __syncthreads();

    if (threadIdx.x == 0) {
        unsigned int ti = 0, tu = 0;
#pragma unroll
        for (int w = 0; w < THREADS / 32; ++w) { ti += sI[w]; tu += sU[w]; }
        atomicAdd(&interCnt[b], ti);
        atomicAdd(&unionCnt[b], tu);
    }
}

// One wave32 = exactly B=32 lanes: per-batch IoU then mean.
__global__ void iou_finalize_kernel(const unsigned int* __restrict__ interCnt,
                                    const unsigned int* __restrict__ unionCnt,
                                    float* __restrict__ out) {
    const int b = threadIdx.x;  // 0..31
    float in = (float)interCnt[b];
    float un = (float)unionCnt[b];
    float iou = (un > 0.0f) ? (in / fmaxf(un, 1.0f)) : 0.0f;

#pragma unroll
    for (int off = 16; off > 0; off >>= 1)
        iou += __shfl_xor(iou, off, 32);

    if (b == 0) out[0] = iou * (1.0f / (float)B_SIZE);
}

extern "C" void kernel_launch(void* const* d_in, const int* in_sizes, int n_in,
                              void* d_out, int out_size, void* d_ws, size_t ws_size,
                              hipStream_t stream) {
    const float* pred = (const float*)d_in[0];
    const float* targ = (const float*)d_in[1];
    float* out = (float*)d_out;

    const long long total = (long long)in_sizes[0];   // B*C*H*W
    const int hw = (int)(total / (B_SIZE * NCH));     // 1024*1024
    const int q  = hw / 4;                            // float4 per channel per batch

    unsigned int* cnt = (unsigned int*)d_ws;          // [0..31]=inter, [32..63]=union

    iou_zero_ws<<<1, 64, 0, stream>>>(cnt);

    dim3 grid(BLOCKS_PER_BATCH, B_SIZE);
    iou_count_async<<<grid, THREADS, 0, stream>>>(pred, targ, cnt, cnt + B_SIZE, q);

    iou_finalize_kernel<<<1, 32, 0, stream>>>(cnt, cnt + B_SIZE, out);
}